// FP_Layer_42099269435600
// MI455X (gfx1250) — compile-verified
//
#include <hip/hip_runtime.h>
#include <hip/hip_bf16.h>

// ---------------- problem constants (from reference setup_inputs) ------------
#define BB    8
#define NN    8192
#define MM    2048
#define C1C   128
#define C2C   256
#define CIN   384            // C1 + C2
#define HH    256            // hidden width of both MLP layers
#define NT    (BB * NN)      // 65536 rows
#define NROWBLK (NT / 128)   // 512 GEMM row tiles
#define KC    16             // K-slab per pipeline stage
#define LDT   20             // padded LDS tile stride (floats)

typedef __attribute__((ext_vector_type(2))) float v2f;
typedef __attribute__((ext_vector_type(8))) float v8f;

// ---------------- CDNA5 async global->LDS copy (ASYNCcnt path) ---------------
// builtin signature (from hipcc diagnostic): arg0 = v4i addrspace(1)* (global
// src; clang prints as1 as "__device__"), arg1 = v4i addrspace(3)* (LDS dst),
// then two integer immediates (offset, cpol).
typedef __attribute__((__vector_size__(16))) int v4i_raw;
typedef __attribute__((address_space(1))) v4i_raw g_v4i;
typedef __attribute__((address_space(3))) v4i_raw l_v4i;

static __device__ __forceinline__ void async_cp16(const void* g, void* l) {
#if __has_builtin(__builtin_amdgcn_global_load_async_to_lds_b128)
    // integer detour: generic LDS address low 32 bits == as3 offset;
    // generic global address == as1 address
    __builtin_amdgcn_global_load_async_to_lds_b128(
        (g_v4i*)(uintptr_t)g,
        (l_v4i*)(unsigned)(uintptr_t)l, 0, 0);
#else
    const unsigned loff = (unsigned)(uintptr_t)l;
    asm volatile("global_load_async_to_lds_b128 %0, %1, off"
                 :: "v"(loff), "v"(g) : "memory");
#endif
}

static __device__ __forceinline__ void wait_async0() {
#if __has_builtin(__builtin_amdgcn_s_wait_asynccnt)
    __builtin_amdgcn_s_wait_asynccnt(0);
    __atomic_signal_fence(__ATOMIC_SEQ_CST);
#else
    asm volatile("s_wait_asynccnt 0x0" ::: "memory");
#endif
}

// =============================================================================
// Kernel 0: 3-NN interpolation + concat  ->  xcat (NT x 384)
//   grid = (B, N/256), block = 256
// =============================================================================
__global__ __launch_bounds__(256) void interp_concat_kernel(
    const float* __restrict__ xyz1,     // (B,N,3)
    const float* __restrict__ xyz2,     // (B,M,3)
    const float* __restrict__ points1,  // (B,N,C1)
    const float* __restrict__ points2,  // (B,M,C2)
    float* __restrict__ xcat)           // (NT, CIN)
{
    __shared__ float sxyz2[MM * 3];     // 24 KB: whole xyz2[b]
    __shared__ int   sidx[256][3];
    __shared__ float swgt[256][3];

    const int b    = blockIdx.x;
    const int tile = blockIdx.y;
    const int t    = threadIdx.x;

    // stage xyz2[b] into LDS (coalesced flat copy)
    const float* x2 = xyz2 + (size_t)b * MM * 3;
    for (int i = t; i < MM * 3; i += 256) sxyz2[i] = x2[i];
    __syncthreads();

    // each thread owns one query point
    const int n = tile * 256 + t;
    const float* q = xyz1 + ((size_t)b * NN + n) * 3;
    const float qx = q[0], qy = q[1], qz = q[2];

    float d0 = 3.0e38f, d1 = 3.0e38f, d2 = 3.0e38f;
    int   i0 = 0, i1 = 0, i2 = 0;
    for (int m = 0; m < MM; ++m) {
        const float dx = qx - sxyz2[m * 3 + 0];
        const float dy = qy - sxyz2[m * 3 + 1];
        const float dz = qz - sxyz2[m * 3 + 2];
        const float d  = dx * dx + dy * dy + dz * dz;
        if (d < d2) {
            if (d < d1) {
                if (d < d0) { d2 = d1; i2 = i1; d1 = d0; i1 = i0; d0 = d; i0 = m; }
                else        { d2 = d1; i2 = i1; d1 = d;  i1 = m; }
            } else          { d2 = d;  i2 = m; }
        }
    }
    const float r0 = 1.0f / (d0 + 1e-8f);
    const float r1 = 1.0f / (d1 + 1e-8f);
    const float r2 = 1.0f / (d2 + 1e-8f);
    const float rs = 1.0f / (r0 + r1 + r2);
    sidx[t][0] = i0; sidx[t][1] = i1; sidx[t][2] = i2;
    swgt[t][0] = r0 * rs; swgt[t][1] = r1 * rs; swgt[t][2] = r2 * rs;
    __syncthreads();

    // cooperative, coalesced gather + blend + concat write
    const float* p2b = points2 + (size_t)b * MM * C2C;
    const float* p1b = points1 + (size_t)b * NN * C1C;
    for (int p = 0; p < 256; ++p) {
        const int np = tile * 256 + p;
        const size_t row = ((size_t)b * NN + np) * CIN;
        const int j0 = sidx[p][0], j1 = sidx[p][1], j2 = sidx[p][2];
        const float w0 = swgt[p][0], w1 = swgt[p][1], w2 = swgt[p][2];
        const float v = w0 * p2b[(size_t)j0 * C2C + t]
                      + w1 * p2b[(size_t)j1 * C2C + t]
                      + w2 * p2b[(size_t)j2 * C2C + t];
        xcat[row + C1C + t] = v;                       // interp channels 128..383
        if (t < C1C)
            xcat[row + t] = p1b[(size_t)np * C1C + t]; // points1 channels 0..127
    }
}

// =============================================================================
// Kernels 1/3: WMMA f32 GEMM  Y = A(NT x K) * W(256 x K)^T + bias
//   - async global->LDS staging, double-buffered K-pipeline (KC=16 slabs)
//   - optional fused BN+ReLU applied to A fragments (layer 2)
//   128x128 tile / block, 8 waves in 4x2, each wave 32x64 via 16x16x4 f32 WMMA
//   grid = (NROWBLK, 2), block = 256
// =============================================================================
template <int K, bool NORM_A>
__global__ __launch_bounds__(256) void gemm_bn_kernel(
    const float* __restrict__ A,        // (NT, K)
    const float* __restrict__ W,        // (256, K) row-major
    const float* __restrict__ bias,     // (256)
    const float* __restrict__ bnScaleA, // per-K scale for A (layer-1 BN), or null
    const float* __restrict__ bnShiftA, // per-K shift, or null
    float* __restrict__ Y,              // (NT, 256)
    float* __restrict__ psum,           // (NROWBLK, 256) partial channel sums
    float* __restrict__ psumsq)         // (NROWBLK, 256) partial channel sumsq
{
    constexpr int NSTG = K / KC;

    __shared__ float As[2][128 * LDT];  // 2 x 10 KB
    __shared__ float Bs[2][128 * LDT];  // 2 x 10 KB, Bs[o][k]
    __shared__ float sSc[HH], sSh[HH];  // BN params for A (layer 2 only)
    __shared__ float red_s[8 * 4 * 32];
    __shared__ float red_q[8 * 4 * 32];

    const int tid  = threadIdx.x;
    const int wave = tid >> 5, lane = tid & 31;
    const int wm = wave & 3, wn = wave >> 2;    // wave grid 4 x 2
    const int hi = lane >> 4, ln = lane & 15;
    const int rowBase = blockIdx.x * 128;
    const int colBase = blockIdx.y * 128;

    if (NORM_A) { sSc[tid] = bnScaleA[tid]; sSh[tid] = bnShiftA[tid]; }

    v8f acc[2][4];
    for (int i = 0; i < 2; ++i)
        for (int j = 0; j < 4; ++j) acc[i][j] = {};

    const float* Arow = A + (size_t)rowBase * K;
    const float* Wt   = W + (size_t)colBase * K;

    // per-thread staging coordinates: 128 rows x 4 float4 per row, 2 per thread
    const int str = tid >> 2;              // row 0..63 (+64 for second slot)
    const int stk = (tid & 3) << 2;        // k offset 0,4,8,12

    auto stage = [&](int ks, int buf) {
        const int k0 = ks * KC;
#pragma unroll
        for (int p = 0; p < 2; ++p) {
            const int r = p * 64 + str;
            async_cp16(Arow + (size_t)r * K + k0 + stk, &As[buf][r * LDT + stk]);
            async_cp16(Wt   + (size_t)r * K + k0 + stk, &Bs[buf][r * LDT + stk]);
        }
    };

    stage(0, 0);
    wait_async0();
    __syncthreads();

    for (int ks = 0; ks < NSTG; ++ks) {
        const int cur = ks & 1;
        if (ks + 1 < NSTG) stage(ks + 1, cur ^ 1);   // overlap with compute

        const float* as = As[cur];
        const float* bs = Bs[cur];
        const int m0 = wm * 32, nb = wn * 64;
#pragma unroll
        for (int kk = 0; kk < KC; kk += 4) {
            const int ak = kk + 2 * hi;   // lanes 16..31 hold K=+2,+3 (ISA layout)
            v2f afr[2], bfr[4];
            afr[0] = *(const v2f*)(as + (m0 + ln)      * LDT + ak);
            afr[1] = *(const v2f*)(as + (m0 + 16 + ln) * LDT + ak);
            if (NORM_A) {               // fused BN1+ReLU on A fragments
                const int kg = ks * KC + ak;
                const float sc0 = sSc[kg],     sh0 = sSh[kg];
                const float sc1 = sSc[kg + 1], sh1 = sSh[kg + 1];
#pragma unroll
                for (int i = 0; i < 2; ++i) {
                    afr[i].x = fmaxf(0.f, afr[i].x * sc0 + sh0);
                    afr[i].y = fmaxf(0.f, afr[i].y * sc1 + sh1);
                }
            }
#pragma unroll
            for (int j = 0; j < 4; ++j)
                bfr[j] = *(const v2f*)(bs + (nb + j * 16 + ln) * LDT + ak);
#pragma unroll
            for (int i = 0; i < 2; ++i)
#pragma unroll
                for (int j = 0; j < 4; ++j)
                    acc[i][j] = __builtin_amdgcn_wmma_f32_16x16x4_f32(
                        false, afr[i], false, bfr[j],
                        (short)0, acc[i][j], false, false);
        }
        wait_async0();      // next slab resident in LDS
        __syncthreads();    // everyone done reading cur before it is overwritten
    }

    // ---- epilogue: bias, store, per-channel partial stats (deterministic) ---
#pragma unroll
    for (int j = 0; j < 4; ++j) {
        const int col = colBase + wn * 64 + j * 16 + ln;
        const float bv = bias[col];
        float s = 0.f, sq = 0.f;
#pragma unroll
        for (int i = 0; i < 2; ++i) {
            const int rb = rowBase + wm * 32 + i * 16 + hi * 8;
            const v8f a8 = acc[i][j];
#pragma unroll
            for (int r = 0; r < 8; ++r) {
                const float val = a8[r] + bv;
                Y[(size_t)(rb + r) * HH + col] = val;
                s += val; sq += val * val;
            }
        }
        red_s[(wave * 4 + j) * 32 + lane] = s;
        red_q[(wave * 4 + j) * 32 + lane] = sq;
    }
    __syncthreads();

    if (tid < 128) {   // one thread per tile column, fixed-order reduction
        const int wn2 = tid >> 6, rem = tid & 63;
        const int j = rem >> 4, l = rem & 15;
        float s = 0.f, sq = 0.f;
#pragma unroll
        for (int wm2 = 0; wm2 < 4; ++wm2) {
            const int w = wm2 | (wn2 << 2);
            s  += red_s[(w * 4 + j) * 32 + l] + red_s[(w * 4 + j) * 32 + l + 16];
            sq += red_q[(w * 4 + j) * 32 + l] + red_q[(w * 4 + j) * 32 + l + 16];
        }
        psum  [(size_t)blockIdx.x * HH + colBase + tid] = s;
        psumsq[(size_t)blockIdx.x * HH + colBase + tid] = sq;
    }
}

// =============================================================================
// Kernels 2/4: reduce partials -> per-channel BN scale/shift (1 block, 256 thr)
// =============================================================================
__global__ __launch_bounds__(256) void bn_finalize_kernel(
    const float* __restrict__ psum, const float* __restrict__ psumsq,
    const float* __restrict__ gamma, const float* __restrict__ beta,
    float* __restrict__ scale, float* __restrict__ shift)
{
    const int ch = threadIdx.x;
    float s = 0.f, sq = 0.f;
    for (int r = 0; r < NROWBLK; ++r) {   // fixed order => deterministic
        s  += psum  [(size_t)r * HH + ch];
        sq += psumsq[(size_t)r * HH + ch];
    }
    const float invN = 1.0f / (float)NT;
    const float mean = s * invN;
    const float var  = sq * invN - mean * mean;
    const float sc   = gamma[ch] * rsqrtf(var + 1e-5f);
    scale[ch] = sc;
    shift[ch] = beta[ch] - mean * sc;
}

// =============================================================================
// Kernel 5: in-place BN + ReLU on the final output
// =============================================================================
__global__ __launch_bounds__(256) void bn_relu_kernel(
    float* __restrict__ Y, const float* __restrict__ scale,
    const float* __restrict__ shift)
{
    const size_t i = (size_t)blockIdx.x * 256 + threadIdx.x;
    const int ch = threadIdx.x & (HH - 1);   // blockDim == HH == 256
    Y[i] = fmaxf(0.f, Y[i] * scale[ch] + shift[ch]);
}

// =============================================================================
extern "C" void kernel_launch(void* const* d_in, const int* in_sizes, int n_in,
                              void* d_out, int out_size, void* d_ws, size_t ws_size,
                              hipStream_t stream) {
    (void)in_sizes; (void)n_in; (void)out_size; (void)ws_size;
    const float* xyz1    = (const float*)d_in[0];
    const float* xyz2    = (const float*)d_in[1];
    const float* points1 = (const float*)d_in[2];
    const float* points2 = (const float*)d_in[3];
    const float* W1      = (const float*)d_in[4];
    const float* b1      = (const float*)d_in[5];
    const float* gamma1  = (const float*)d_in[6];
    const float* beta1   = (const float*)d_in[7];
    const float* W2      = (const float*)d_in[8];
    const float* b2      = (const float*)d_in[9];
    const float* gamma2  = (const float*)d_in[10];
    const float* beta2   = (const float*)d_in[11];
    float* out = (float*)d_out;

    // workspace carve-up
    float* xcat    = (float*)d_ws;                       // NT*384
    float* y1      = xcat    + (size_t)NT * CIN;         // NT*256
    float* psum1   = y1      + (size_t)NT * HH;          // NROWBLK*256
    float* psumsq1 = psum1   + (size_t)NROWBLK * HH;
    float* psum2   = psumsq1 + (size_t)NROWBLK * HH;
    float* psumsq2 = psum2   + (size_t)NROWBLK * HH;
    float* scale1  = psumsq2 + (size_t)NROWBLK * HH;     // 256
    float* shift1  = scale1 + HH;
    float* scale2  = shift1 + HH;
    float* shift2  = scale2 + HH;

    // 0) 3-NN interpolation + concat
    interp_concat_kernel<<<dim3(BB, NN / 256), 256, 0, stream>>>(
        xyz1, xyz2, points1, points2, xcat);

    // 1) GEMM1 + bias, partial BN stats
    gemm_bn_kernel<CIN, false><<<dim3(NROWBLK, 2), 256, 0, stream>>>(
        xcat, W1, b1, nullptr, nullptr, y1, psum1, psumsq1);
    bn_finalize_kernel<<<1, HH, 0, stream>>>(psum1, psumsq1, gamma1, beta1,
                                             scale1, shift1);

    // 2) GEMM2 with fused BN1+ReLU on A, writes raw y2 into d_out
    gemm_bn_kernel<HH, true><<<dim3(NROWBLK, 2), 256, 0, stream>>>(
        y1, W2, b2, scale1, shift1, out, psum2, psumsq2);
    bn_finalize_kernel<<<1, HH, 0, stream>>>(psum2, psumsq2, gamma2, beta2,
                                             scale2, shift2);

    // 3) in-place BN2 + ReLU
    bn_relu_kernel<<<NT * HH / 256, 256, 0, stream>>>(out, scale2, shift2);
}